// DMPNNConv_42399917146352
// MI455X (gfx1250) — compile-verified
//
#include <hip/hip_runtime.h>

// D-MPNN conv for MI455X (gfx1250, wave32).
// GEMMs run on V_WMMA_F32_16X16X4_F32 (full fp32 precision, matrix pipe).
// Memory-bound workload (~2.5-3 GB traffic vs 27 GFLOP), so layout/coalescing
// and atomic scatter efficiency dominate; WMMA offloads the 128->64 projections.

typedef __attribute__((ext_vector_type(2))) float v2f;
typedef __attribute__((ext_vector_type(8))) float v8f;

#define HID 64
#define IN_DIM 128

// ---------------------------------------------------------------------------
// Kernel 1: in_sum[row[e]] += edge_hidden[e]   (segment_sum over source node)
// ---------------------------------------------------------------------------
__global__ __launch_bounds__(256) void scatter_in_sum_kernel(
    const float* __restrict__ eh, const int* __restrict__ row,
    float* __restrict__ in_sum, int total /* = E*HID */) {
  int idx = blockIdx.x * 256 + threadIdx.x;
  if (idx >= total) return;
  int e = idx >> 6;           // edge
  int f = idx & 63;           // feature
  atomicAdd(&in_sum[(size_t)row[e] * HID + f], eh[idx]);
}

// ---------------------------------------------------------------------------
// Kernel 2/3: wave-level 16x16-tile GEMM, K=128, fp32 WMMA.
//
// Each wave owns one N-tile (16 output cols) and holds all 32 K-group
// B-fragments in registers (64 VGPRs), then grid-strides over M-tiles.
//
// MSG=true : A row m = [ x[row[e0+m]] , in_sum[col[e0+m]] - eh[(e0+m)^1] ]
//            out = messages[E,64]; also atomic-scatter into node_msg by col.
// MSG=false: A row m = [ x[m0+m] , node_msg[m0+m] ] ; out = x_out[N,64].
// ---------------------------------------------------------------------------
template <bool MSG>
__global__ __launch_bounds__(256) void wave_gemm_kernel(
    const float* __restrict__ left,     // x [N,64]
    const float* __restrict__ right,    // in_sum (MSG) or node_msg (!MSG) [N,64]
    const float* __restrict__ eh,       // edge_hidden [E,64]       (MSG only)
    const int* __restrict__ row,        // edge_index[0]            (MSG only)
    const int* __restrict__ col,        // edge_index[1]            (MSG only)
    const float* __restrict__ W,        // [64][128] row-major (out x in)
    const float* __restrict__ bias,     // [64]
    float* __restrict__ out,            // messages [E,64] or x_out [N,64]
    float* __restrict__ scatter,        // node_msg [N,64]          (MSG only)
    int n_mtiles, int mtile_stride) {
  const int lane = threadIdx.x & 31;
  const int wave = blockIdx.x * (blockDim.x >> 5) + (threadIdx.x >> 5);
  const int ntile = wave & 3;          // which 16 of the 64 output cols
  const int mtile0 = wave >> 2;
  const int half = lane >> 4;          // 0: K{4k,4k+1}, 1: K{4k+2,4k+3}
  const int lidx = lane & 15;
  const int n = ntile * 16 + lidx;     // output column this lane produces

  // --- Preload B fragments: B[k][n] = W[n][k]; lane needs (n, 4k+2*half..+1).
  v2f barr[32];
  const float* wrow = W + (size_t)n * IN_DIM + 2 * half;
#pragma unroll
  for (int k = 0; k < 32; ++k) barr[k] = *(const v2f*)(wrow + 4 * k);
  const float bb = bias[n];

  for (int mtile = mtile0; mtile < n_mtiles; mtile += mtile_stride) {
    const int m0 = mtile * 16;
    const int em = m0 + lidx;          // the A-row this lane feeds
    const float* pL;
    const float* pR;
    const float* pE = nullptr;
    if constexpr (MSG) {
      pL = left + (size_t)row[em] * HID;          // x[row[e]]
      pR = right + (size_t)col[em] * HID;         // in_sum[col[e]]
      pE = eh + (size_t)(em ^ 1) * HID;           // edge_hidden[rev[e]]
    } else {
      pL = left + (size_t)em * HID;               // x[node]
      pR = right + (size_t)em * HID;              // node_msg[node]
    }

    v8f acc = {};
    const int off = 2 * half;
    // K = 0..63 : left half of the concatenated input
#pragma unroll
    for (int k = 0; k < 16; ++k) {
      v2f a = *(const v2f*)(pL + 4 * k + off);
      acc = __builtin_amdgcn_wmma_f32_16x16x4_f32(
          false, a, false, barr[k], (short)0, acc, false, false);
    }
    // K = 64..127 : right half (minus reverse-edge hidden for MSG)
#pragma unroll
    for (int k = 0; k < 16; ++k) {
      v2f a = *(const v2f*)(pR + 4 * k + off);
      if constexpr (MSG) {
        v2f s = *(const v2f*)(pE + 4 * k + off);
        a.x -= s.x;
        a.y -= s.y;
      }
      acc = __builtin_amdgcn_wmma_f32_16x16x4_f32(
          false, a, false, barr[16 + k], (short)0, acc, false, false);
    }

    // Epilogue: D layout — VGPR r holds row (r + 8*half), col = lidx.
#pragma unroll
    for (int r8 = 0; r8 < 8; ++r8) {
      const int m = m0 + r8 + 8 * half;
      float v = acc[r8] + bb;
      v = v > 0.f ? v : 0.f;
      out[(size_t)m * HID + n] = v;
      if constexpr (MSG) {
        atomicAdd(&scatter[(size_t)col[m] * HID + n], v);
      }
    }
  }
}

// ---------------------------------------------------------------------------
// Launch
// ---------------------------------------------------------------------------
extern "C" void kernel_launch(void* const* d_in, const int* in_sizes, int n_in,
                              void* d_out, int out_size, void* d_ws,
                              size_t ws_size, hipStream_t stream) {
  const float* x = (const float*)d_in[0];       // [N,64]
  const int* edge_index = (const int*)d_in[1];  // [2,E]
  const float* eh = (const float*)d_in[2];      // [E,64]
  const float* W_msg = (const float*)d_in[3];   // [64,128]
  const float* b_msg = (const float*)d_in[4];   // [64]
  const float* W_node = (const float*)d_in[5];  // [64,128]
  const float* b_node = (const float*)d_in[6];  // [64]

  const int N = in_sizes[0] / HID;
  const int E = in_sizes[1] / 2;
  const int* row = edge_index;
  const int* col = edge_index + E;

  float* x_out = (float*)d_out;                    // [N,64]
  float* messages = x_out + (size_t)N * HID;       // [E,64]

  float* in_sum = (float*)d_ws;                    // [N,64]
  float* node_msg = in_sum + (size_t)N * HID;      // [N,64]

  // Zero both segment-sum accumulators (graph-capture safe).
  hipMemsetAsync(d_ws, 0, 2 * (size_t)N * HID * sizeof(float), stream);

  // 1) in_sum = segment_sum(edge_hidden, row)
  {
    int total = E * HID;
    int blocks = (total + 255) / 256;
    scatter_in_sum_kernel<<<blocks, 256, 0, stream>>>(eh, row, in_sum, total);
  }

  // 2) messages GEMM + node_msg scatter. E = 1.6M -> 100000 M-tiles of 16.
  {
    const int n_mtiles = (E + 15) / 16;
    const int blocks = 2048;                 // 16384 waves, persistent B regs
    const int mstride = blocks * 8 / 4;      // waves / 4 N-tiles
    wave_gemm_kernel<true><<<blocks, 256, 0, stream>>>(
        x, in_sum, eh, row, col, W_msg, b_msg, messages, node_msg, n_mtiles,
        mstride);
  }

  // 3) x_out GEMM. N = 50000 -> 3125 M-tiles, direct map (one tile per wave).
  {
    const int n_mtiles = (N + 15) / 16;
    const int waves = n_mtiles * 4;
    const int blocks = (waves + 7) / 8;
    wave_gemm_kernel<false><<<blocks, 256, 0, stream>>>(
        x, node_msg, nullptr, nullptr, nullptr, W_node, b_node, x_out, nullptr,
        n_mtiles, 1 << 30);
  }
}